// CTCLoss_1116691497597
// MI455X (gfx1250) — compile-verified
//
#include <hip/hip_runtime.h>
#include <hip/hip_bf16.h>
#include <math.h>

typedef __attribute__((ext_vector_type(2))) float v2f;
typedef __attribute__((ext_vector_type(8))) float v8f;

#define NEGF (-1e30f)
#define VDIM 128

// ---------------------------------------------------------------------------
// Kernel 1: lse[row] = logsumexp over the 128 logits of each (b,t) row.
// One wave handles 16 rows. Lane L and lane L+16 pair on row (base+L):
//   low lane loads v = 4c, 4c+1 ; high lane loads v = 4c+2, 4c+3  (c = 0..31)
// which is exactly the V_WMMA_F32_16X16X4_F32 A-matrix layout (lanes 0-15:
// K0/K1 in VGPR0/1, lanes 16-31: K2/K3). B = all-ones => D[m,n] = rowsum(m).
// Hardware exp (v_exp_f32) is safe: argument is <= 0 and bounded.
// ---------------------------------------------------------------------------
__global__ void __launch_bounds__(128)
lse_kernel(const float* __restrict__ logits, float* __restrict__ lse, int rows)
{
    const int lane   = threadIdx.x & 31;
    const int waveId = (blockIdx.x * blockDim.x + threadIdx.x) >> 5;
    const int rowBase = waveId * 16;
    if (rowBase >= rows) return;                 // uniform per wave

    const bool hi   = lane >= 16;
    const int  myRow = rowBase + (lane & 15);
    const bool valid = myRow < rows;
    const int  ldRow = valid ? myRow : rowBase;  // clamp (keeps WMMA uniform)

    const float* rp = logits + (size_t)ldRow * VDIM + (hi ? 2 : 0);

    // Load this lane's 64 values (v ≡ {0,1} or {2,3} mod 4) into registers.
    float vals[64];
#pragma unroll
    for (int c = 0; c < 32; ++c) {
        float2 v = *(const float2*)(rp + 4 * c);
        vals[2 * c]     = v.x;
        vals[2 * c + 1] = v.y;
    }

    // Row max: local 64-way max, then combine with the paired lane.
    float m = vals[0];
#pragma unroll
    for (int i = 1; i < 64; ++i) m = fmaxf(m, vals[i]);
    m = fmaxf(m, __shfl_xor(m, 16));

    // Sum of exp via f32 WMMA with B = ones: 32 K-chunks of 4.
    v8f acc = {0.f, 0.f, 0.f, 0.f, 0.f, 0.f, 0.f, 0.f};
    const v2f ones = {1.0f, 1.0f};
#pragma unroll
    for (int c = 0; c < 32; ++c) {
        v2f a;
        a.x = __expf(vals[2 * c]     - m);   // v_exp_f32, arg in [-inf, 0]
        a.y = __expf(vals[2 * c + 1] - m);
        acc = __builtin_amdgcn_wmma_f32_16x16x4_f32(
                  false, a, false, ones, (short)0, acc, false, false);
    }

    // D layout: lanes 0-15: acc[j] = rowsum(j); lanes 16-31: acc[j] = rowsum(j+8).
    // Lane wants rowsum(lane & 15). Select acc[lane&7], then lanes 8..23 take
    // the paired lane's value.
    const int r = lane & 7;
    float s01 = (r & 1) ? acc[1] : acc[0];
    float s23 = (r & 1) ? acc[3] : acc[2];
    float s45 = (r & 1) ? acc[5] : acc[4];
    float s67 = (r & 1) ? acc[7] : acc[6];
    float s03 = (r & 2) ? s23 : s01;
    float s47 = (r & 2) ? s67 : s45;
    float sel = (r & 4) ? s47 : s03;
    float other = __shfl_xor(sel, 16);
    int grp = (lane >> 3) & 3;                 // 1,2 => need paired lane's value
    float rowsum = (grp == 1 || grp == 2) ? other : sel;

    float lseVal = m + __logf(rowsum);         // rowsum in [1, 128]
    if (valid && !hi) lse[myRow] = lseVal;     // lanes 0..15 -> 16 consecutive floats
}

// ---------------------------------------------------------------------------
// Kernel 2: CTC forward recursion. One block per batch element b.
// Thread s owns extended-label slot s (L = 2S+1 <= 256). Alpha ping-pong and
// the current logit row live in LDS; the next row is double-buffered and
// stored before the single per-step barrier. Hardware transcendentals keep
// the serial dependency chain per step short (2x v_exp + 2x v_log).
// ---------------------------------------------------------------------------
__device__ __forceinline__ float logaddexpf_(float a, float b)
{
    float mx = fmaxf(a, b);
    // |a-b| huge -> __expf underflows to 0 -> __logf(1) = 0 -> result = mx.
    return mx + __logf(1.0f + __expf(-fabsf(a - b)));
}

__global__ void __launch_bounds__(256)
ctc_dp_kernel(const float* __restrict__ logits,
              const int*   __restrict__ targets,
              const int*   __restrict__ in_lens,
              const int*   __restrict__ tgt_lens,
              const float* __restrict__ lse,
              float*       __restrict__ losses,
              int T, int S)
{
    __shared__ float rowbuf[2][VDIM];
    __shared__ float alpha[2][256];

    const int b   = blockIdx.x;
    const int tid = threadIdx.x;
    const int L   = 2 * S + 1;
    const int Tb  = in_lens[b];
    const int U   = tgt_lens[b];

    const float* lrow = logits + (size_t)b * T * VDIM;  // [T][128]
    const float* lseb = lse + (size_t)b * T;

    // Per-slot label index and skip flag.
    const bool active = (tid < L) && (tid < 256);
    int  myE = 0;
    bool mySkip = false;
    if (active && (tid & 1)) {
        int k = tid >> 1;                       // 0-based target index
        myE = targets[b * S + k];
        mySkip = (k >= 1) && (myE != 0) && (myE != targets[b * S + k - 1]);
    }

    // Load rows 0 and 1 into the double buffer.
    if (tid < 32) {
        *(float4*)&rowbuf[0][tid * 4] = *(const float4*)(lrow + tid * 4);
        if (Tb > 1)
            *(float4*)&rowbuf[1][tid * 4] = *(const float4*)(lrow + VDIM + tid * 4);
    }
    __syncthreads();

    // alpha at t = 0.
    {
        float lse0 = lseb[0];
        float a = NEGF;
        if (tid == 0)      a = rowbuf[0][0]   - lse0;
        else if (tid == 1) a = (U > 0) ? (rowbuf[0][myE] - lse0) : NEGF;
        if (active) alpha[0][tid] = a;
    }
    __syncthreads();

    int cur = 0;
    for (int t = 1; t < Tb; ++t) {
        const int rb = t & 1;                   // row t lives in rowbuf[rb]

        // Warm rows further ahead into cache (gfx1250 global_prefetch_b8).
        if (tid < 32 && (t + 4) < Tb)
            __builtin_prefetch(lrow + (size_t)(t + 4) * VDIM + tid * 4, 0, 1);

        // Issue the t+1 row load early; store to LDS after the compute.
        float4 pf;
        const bool doPf = (tid < 32) && (t + 1 < Tb);
        if (doPf) pf = *(const float4*)(lrow + (size_t)(t + 1) * VDIM + tid * 4);

        const float lst = lseb[t];              // uniform scalar

        float na = NEGF;
        if (active) {
            float a0 = alpha[cur][tid];
            float a1 = (tid >= 1) ? alpha[cur][tid - 1] : NEGF;
            float a2 = mySkip ? alpha[cur][tid - 2] : NEGF;
            float s  = logaddexpf_(logaddexpf_(a0, a1), a2);
            na = s + (rowbuf[rb][myE] - lst);
        }

        if (doPf) *(float4*)&rowbuf[rb ^ 1][tid * 4] = pf;
        if (active) alpha[cur ^ 1][tid] = na;
        __syncthreads();
        cur ^= 1;
    }

    if (tid == 0) {
        const float* A = alpha[cur];
        int idx = 2 * U;
        float lastBlank = A[idx];
        float lastLabel = (U > 0) ? A[idx - 1] : NEGF;
        losses[b] = -logaddexpf_(lastBlank, lastLabel);
    }
}

// ---------------------------------------------------------------------------
// Kernel 3: deterministic sum of per-batch losses -> scalar output.
// ---------------------------------------------------------------------------
__global__ void reduce_loss_kernel(const float* __restrict__ losses,
                                   float* __restrict__ out, int B)
{
    if (threadIdx.x == 0 && blockIdx.x == 0) {
        float acc = 0.f;
        for (int i = 0; i < B; ++i) acc += losses[i];
        out[0] = acc;
    }
}

extern "C" void kernel_launch(void* const* d_in, const int* in_sizes, int n_in,
                              void* d_out, int out_size, void* d_ws, size_t ws_size,
                              hipStream_t stream)
{
    const float* logits   = (const float*)d_in[0];
    const int*   targets  = (const int*)d_in[1];
    const int*   in_lens  = (const int*)d_in[2];
    const int*   tgt_lens = (const int*)d_in[3];

    const int B = in_sizes[2];
    const int S = in_sizes[1] / B;
    const int T = in_sizes[0] / (B * VDIM);
    const int rows = B * T;

    float* lse    = (float*)d_ws;          // B*T floats
    float* losses = lse + (size_t)rows;    // B floats
    float* out    = (float*)d_out;

    // Kernel 1: 128 threads = 4 waves/block, 16 rows/wave => 64 rows/block.
    int blocks1 = (rows + 63) / 64;
    lse_kernel<<<blocks1, 128, 0, stream>>>(logits, lse, rows);

    // Kernel 2: one block per batch element.
    ctc_dp_kernel<<<B, 256, 0, stream>>>(logits, targets, in_lens, tgt_lens,
                                         lse, losses, T, S);

    // Kernel 3: final scalar.
    reduce_loss_kernel<<<1, 32, 0, stream>>>(losses, out, B);
}